// FamilyMLPLinear_26843545600586
// MI455X (gfx1250) — compile-verified
//
#include <hip/hip_runtime.h>
#include <hip/hip_bf16.h>

typedef __attribute__((ext_vector_type(2))) float v2f;
typedef __attribute__((ext_vector_type(8))) float v8f;

constexpr int NM = 16;      // num models (contraction length per class)
constexpr int NB = 512;     // batch
constexpr int NC = 17929;   // classes
constexpr int NH = 4;       // hidden

constexpr int BT = 32;      // batch rows per workgroup (two 16-row WMMA subtiles)
constexpr int CT = 32;      // classes per workgroup (4 per wave)
constexpr int PM = 40;      // m-stride inside a class slab of xs (floats)
constexpr int PC = 16 * PM + 1;   // = 641: class pitch of xs; odd -> conflict-free async stores
constexpr int W1P = 67;     // class pitch of w1s (conflict-free B reads)

// Async GVS-form memory->LDS copy: saddr = uniform SGPR base, voffset = per-lane VGPR,
// imm offset applies to BOTH the LDS destination and the global address (ISA 08 §4.4).
// IB*4 walks consecutive batch rows on the LDS side; the global side's row stride NC*4
// is carried by voff, pre-compensated by -IB*4 (hence the (NC-1)*4 update).
template<int IB>
struct XStage {
    static __device__ __forceinline__ void run(unsigned ldst, unsigned voff, const float* sbase) {
        asm volatile("global_load_async_to_lds_b32 %0, %1, %2 offset:%c3"
                     :: "v"(ldst), "v"(voff), "s"(sbase), "n"(IB * 4)
                     : "memory");
        if constexpr (IB + 1 < 32)
            XStage<IB + 1>::run(ldst, voff + (unsigned)((NC - 1) * 4), sbase);
    }
};

__device__ __forceinline__ void wait_asynccnt0() {
    asm volatile("s_wait_asynccnt 0" ::: "memory");
}

__global__ __launch_bounds__(256)
void fam_mlp_kernel(const float* __restrict__ x,    // [NM, NB, NC]
                    const float* __restrict__ W1,   // [NC, NH, NM]
                    const float* __restrict__ b1,   // [NC, NH]
                    const float* __restrict__ W2,   // [NC, NH]
                    const float* __restrict__ b2,   // [NC]
                    float* __restrict__ out) {      // [NB, NC]
    __shared__ float xs[CT * PC];     // xs[c_local][m*PM + b2]
    __shared__ float w1s[CT * W1P];   // w1s[c_local][h*16 + m]

    const int tid  = threadIdx.x;
    const int wave = tid >> 5;
    const int lane = tid & 31;
    const int c0 = blockIdx.x * CT;
    const int b0 = blockIdx.y * BT;

    // ---- stage x tile: coalesced 128B rows (lanes over c), transposed scatter into
    //      LDS by the async unit. 1 v_add + 1 async op per transfer. ----
    {
        int c = c0 + lane;
        if (c >= NC) c = NC - 1;                    // clamp; bogus columns never stored
        const unsigned c4 = (unsigned)c * 4u;
        const unsigned xs_lane = (unsigned)(uintptr_t)(&xs[0]) + (unsigned)(lane * PC * 4);
        #pragma unroll
        for (int mm = 0; mm < 2; ++mm) {
            const int m = wave * 2 + mm;            // 8 waves x 2 = all 16 m's
            const unsigned ldst  = xs_lane + (unsigned)(m * PM * 4);
            const unsigned voff0 = (unsigned)(((size_t)m * NB + (size_t)b0) * NC * 4u) + c4;
            XStage<0>::run(ldst, voff0, x);         // 32 batch rows
        }
    }
    // ---- stage W1 tile (contiguous 2048-float global region) ----
    {
        const unsigned w1s_base = (unsigned)(uintptr_t)(&w1s[0]);
        #pragma unroll
        for (int i = 0; i < 8; ++i) {
            const int idx = i * 256 + tid;          // 0..2047
            const int cl  = idx >> 6;
            const int rem = idx & 63;               // h*16 + m
            int c = c0 + cl;
            if (c >= NC) c = NC - 1;
            const unsigned voff = (unsigned)((c * (NH * NM) + rem) * 4);
            const unsigned ldst = w1s_base + (unsigned)((cl * W1P + rem) * 4);
            asm volatile("global_load_async_to_lds_b32 %0, %1, %2"
                         :: "v"(ldst), "v"(voff), "s"(W1) : "memory");
        }
    }
    wait_asynccnt0();
    __syncthreads();

    // Column mapping for D: j = csub*4 + hh  (csub = class-within-wave, hh = hidden)
    const int bl   = lane & 15;     // A row (batch) / B,C,D column index
    const int half = lane >> 4;
    const int csub = bl >> 2;
    const int hh   = bl & 3;
    const int cw   = wave * 4;      // first class-within-WG owned by this wave

    const int  cg     = c0 + cw + csub;     // this lane's class
    const bool cvalid = (cg < NC);
    const int  cidx   = cvalid ? cg : (NC - 1);
    const float bias1 = b1[cidx * NH + hh];
    const float w2v   = W2[cidx * NH + hh];
    const float bias2 = b2[cidx];

    // ---- build the 16 block-diagonal B operands ONCE (shared by both subtiles) ----
    // B_t[k, j]: zero unless this column's class (csub) == t's class block (t>>2).
    v2f bops[16];
    {
        const float* wsrc = &w1s[(cw + csub) * W1P + hh * 16];
        #pragma unroll
        for (int t = 0; t < 16; ++t) {
            const int tc = t >> 2;
            const int m0 = ((t & 3) << 2) + half * 2;    // this lane's first k (m)
            const float gate = (csub == tc) ? 1.0f : 0.0f;
            bops[t].x = gate * wsrc[m0];
            bops[t].y = gate * wsrc[m0 + 1];
        }
    }

    #pragma unroll
    for (int bsub = 0; bsub < 2; ++bsub) {
        // b1 folded into the accumulator init (C operand of the first WMMA)
        v8f acc = {bias1, bias1, bias1, bias1, bias1, bias1, bias1, bias1};

        // 16 chained f32 WMMAs: k blocks = (class_sub 0..3) x (m in groups of 4)
        #pragma unroll
        for (int t = 0; t < 16; ++t) {
            const int tc = t >> 2;                       // class block this WMMA contracts
            const int m0 = ((t & 3) << 2) + half * 2;    // this lane's first k (m)
            v2f a;
            // A[b, k]: x[m, b0 + bsub*16 + b, c0+cw+tc]
            const float* xsrc = &xs[(cw + tc) * PC + m0 * PM + bsub * 16 + bl];
            a.x = xsrc[0];
            a.y = xsrc[PM];
            acc = __builtin_amdgcn_wmma_f32_16x16x4_f32(false, a, false, bops[t],
                                                        (short)0, acc, false, false);
        }

        // ---- epilogue: ReLU, *W2, reduce over hh, +b2, store ----
        float o[8];
        #pragma unroll
        for (int r = 0; r < 8; ++r)
            o[r] = fmaxf(acc[r], 0.0f) * w2v;
        #pragma unroll
        for (int r = 0; r < 8; ++r) {
            o[r] += __shfl_xor(o[r], 1, 32);
            o[r] += __shfl_xor(o[r], 2, 32);
        }
        if (hh == 0 && cvalid) {
            // lanes {0,4,8,12}(+16) hold 4 consecutive classes for the same rows -> 16B chunks
            #pragma unroll
            for (int r = 0; r < 8; ++r) {
                const int brow = b0 + bsub * 16 + r + half * 8;   // D row = r + 8*half
                out[(size_t)brow * NC + cg] = o[r] + bias2;
            }
        }
    }
}

extern "C" void kernel_launch(void* const* d_in, const int* in_sizes, int n_in,
                              void* d_out, int out_size, void* d_ws, size_t ws_size,
                              hipStream_t stream) {
    const float* x  = (const float*)d_in[0];
    const float* W1 = (const float*)d_in[1];
    const float* b1 = (const float*)d_in[2];
    const float* W2 = (const float*)d_in[3];
    const float* b2 = (const float*)d_in[4];
    float* out = (float*)d_out;

    dim3 grid((NC + CT - 1) / CT, NB / BT);   // 561 x 16
    fam_mlp_kernel<<<grid, 256, 0, stream>>>(x, W1, b1, W2, b2, out);
}